// GNNEncoder_52252572123529
// MI455X (gfx1250) — compile-verified
//
#include <hip/hip_runtime.h>

// GCN 2-layer forward for MI455X (gfx1250, wave32).
// GEMMs via V_WMMA_F32_16X16X4_F32 (exact fp32) with fused epilogue
// (self-loop + bias init of the aggregation buffer) and fused ReLU on the
// layer-2 A-operand loads. Edge aggregation: one wave per edge, float4
// gathers + global_atomic_add_f32 scatters (entire working set is L2-resident:
// h/agg/edge_index ~51 MB each vs 192 MB L2).

typedef __attribute__((ext_vector_type(2))) float v2f;
typedef __attribute__((ext_vector_type(8))) float v8f;

#define D 128

__global__ void k_zero(float* __restrict__ p, int n) {
  int i = blockIdx.x * blockDim.x + threadIdx.x;
  if (i < n) p[i] = 0.0f;
}

__global__ void k_degree(const long long* __restrict__ dst, float* __restrict__ deg, int ne) {
  int i = blockIdx.x * blockDim.x + threadIdx.x;
  if (i < ne) atomicAdd(&deg[(int)dst[i]], 1.0f);
}

__global__ void k_dinv(float* __restrict__ deg, int n) {
  int i = blockIdx.x * blockDim.x + threadIdx.x;
  if (i < n) deg[i] = rsqrtf(deg[i] + 1.0f);  // deg includes self loop (+1)
}

// H[m,n] = sum_k act(X[m,k]) * W[k,n]   (act = ReLU if RELU_IN)
// AGG[m,n] = H[m,n] * dinv[m]^2 + bias[n]   (self-loop + bias, fused epilogue)
//
// One 16x16 C tile per wave; 8 waves/block cover the 128 output columns.
// f32 WMMA layouts (ISA 7.12.2): A 16x4 and B 4x16 each use 2 VGPRs/lane with
// K = kbase + 2*(lane>=16) + vgpr; C/D: VGPR r holds M = r + 8*(lane>=16), N = lane&15.
template <bool RELU_IN>
__global__ void k_gemm_wmma(const float* __restrict__ X, const float* __restrict__ W,
                            const float* __restrict__ dinv, const float* __restrict__ bias,
                            float* __restrict__ H, float* __restrict__ AGG, int nrows) {
  const int wave = threadIdx.x >> 5;
  const int lane = threadIdx.x & 31;
  const int half = lane >> 4;
  const int l    = lane & 15;
  const int m0 = blockIdx.x << 4;
  const int n0 = wave << 4;

  int row  = m0 + l;
  int rrow = row < nrows ? row : nrows - 1;   // clamp reads: EXEC must stay all-1s for WMMA
  const float* __restrict__ xrow = X + (size_t)rrow * D;

  v8f c = {};
  for (int k = 0; k < D; k += 4) {
    const int kk = k + (half << 1);
    v2f a, b;
    a.x = xrow[kk];
    a.y = xrow[kk + 1];
    if (RELU_IN) { a.x = fmaxf(a.x, 0.0f); a.y = fmaxf(a.y, 0.0f); }
    b.x = W[(size_t)kk * D + n0 + l];
    b.y = W[(size_t)(kk + 1) * D + n0 + l];
    // 8 args: (neg_a, A, neg_b, B, c_mod, C, reuse_a, reuse_b)
    c = __builtin_amdgcn_wmma_f32_16x16x4_f32(false, a, false, b, (short)0, c,
                                              false, false);
  }

  const float bn = bias[n0 + l];
  if (m0 + 16 <= nrows) {
    // fast path: full tile, no per-row exec-mask churn
#pragma unroll
    for (int r = 0; r < 8; ++r) {
      const int    orow = m0 + r + (half << 3);
      const size_t idx  = (size_t)orow * D + n0 + l;
      const float  di   = dinv[orow];
      H[idx]   = c[r];
      AGG[idx] = c[r] * (di * di) + bn;
    }
  } else {
#pragma unroll
    for (int r = 0; r < 8; ++r) {
      const int orow = m0 + r + (half << 3);
      if (orow < nrows) {
        const size_t idx = (size_t)orow * D + n0 + l;
        const float  di  = dinv[orow];
        H[idx]   = c[r];
        AGG[idx] = c[r] * (di * di) + bn;
      }
    }
  }
}

// One wave per edge: 32 lanes x float4 = full 128-float row per wave.
// Gather h[src] (coalesced 512B), scatter-add into agg[dst] via f32 atomics.
__global__ void k_edge(const long long* __restrict__ src, const long long* __restrict__ dst,
                       const float* __restrict__ dinv, const float* __restrict__ h,
                       float* __restrict__ agg, int ne) {
  int gw   = (blockIdx.x * blockDim.x + threadIdx.x) >> 5;  // global wave id = edge id
  int lane = threadIdx.x & 31;
  if (gw >= ne) return;

  int s = (int)src[gw];
  int d = (int)dst[gw];
  float norm = dinv[s] * dinv[d];

  const float4* __restrict__ hv = (const float4*)(h + (size_t)s * D);
  float4 v = hv[lane];

  float* a = agg + (size_t)d * D + lane * 4;
  atomicAdd(a + 0, v.x * norm);
  atomicAdd(a + 1, v.y * norm);
  atomicAdd(a + 2, v.z * norm);
  atomicAdd(a + 3, v.w * norm);
}

extern "C" void kernel_launch(void* const* d_in, const int* in_sizes, int n_in,
                              void* d_out, int out_size, void* d_ws, size_t ws_size,
                              hipStream_t stream) {
  const float*     x  = (const float*)d_in[0];
  const long long* ei = (const long long*)d_in[1];   // int64 edge_index [2, E]
  const float*     W1 = (const float*)d_in[2];
  const float*     b1 = (const float*)d_in[3];
  const float*     W2 = (const float*)d_in[4];
  const float*     b2 = (const float*)d_in[5];
  float*           out = (float*)d_out;

  const int nn = in_sizes[0] / D;       // 100000 nodes
  const int ne = in_sizes[1] / 2;       // 3200000 edges
  const long long* src = ei;
  const long long* dst = ei + ne;

  // Workspace: dinv [nn] | bufA (h) [nn*128] | bufB (agg) [nn*128]  (~103 MB)
  char* ws = (char*)d_ws;
  size_t o = 0;
  float* dinv = (float*)(ws + o); o += ((size_t)nn * 4 + 255) & ~(size_t)255;
  float* bufA = (float*)(ws + o); o += (size_t)nn * D * sizeof(float);
  float* bufB = (float*)(ws + o);

  const int T = 256;                    // 8 wave32 per block
  const int gN  = (nn + T - 1) / T;
  const int gE  = (ne + T - 1) / T;
  const int gG  = (nn + 15) / 16;       // one 16-row tile per block
  const int gEw = (ne + 7) / 8;         // one edge per wave, 8 waves per block

  // Degrees -> dinv = rsqrt(deg + 1)
  k_zero  <<<gN, T, 0, stream>>>(dinv, nn);
  k_degree<<<gE, T, 0, stream>>>(dst, dinv, ne);
  k_dinv  <<<gN, T, 0, stream>>>(dinv, nn);

  // Layer 1: bufA = h1 = x @ W1 ; bufB = h1*dinv^2 + b1 (fused) ; edge scatter
  k_gemm_wmma<false><<<gG,  T, 0, stream>>>(x, W1, dinv, b1, bufA, bufB, nn);
  k_edge            <<<gEw, T, 0, stream>>>(src, dst, dinv, bufA, bufB, ne);

  // Layer 2: bufA = h2 = relu(bufB) @ W2 (ReLU fused into A loads) ;
  //          out  = h2*dinv^2 + b2 (fused) ; edge scatter into out
  k_gemm_wmma<true><<<gG,  T, 0, stream>>>(bufB, W2, dinv, b2, bufA, out, nn);
  k_edge           <<<gEw, T, 0, stream>>>(src, dst, dinv, bufA, out, ne);
}